// Actor_decoder_46875273068975
// MI455X (gfx1250) — compile-verified
//
#include <hip/hip_runtime.h>
#include <math.h>

// Problem constants (from reference)
#define BATCH   64
#define NCAND   20
#define KSEL    5
#define FEAT    320
#define MID     128
#define MCOMB   7752            // C(20,5)/2
#define ROWS    (BATCH*NCAND)   // 1280
#define NTILES  ((MCOMB + 15) / 16)  // 485

typedef float v2f __attribute__((ext_vector_type(2)));
typedef float v8f __attribute__((ext_vector_type(8)));

// BINOM[n][k] = C(n,k) for n in 0..19, k in 0..4 (for lex unranking)
__device__ __constant__ int BINOM[20][5] = {
 {1,0,0,0,0},{1,1,0,0,0},{1,2,1,0,0},{1,3,3,1,0},{1,4,6,4,1},
 {1,5,10,10,5},{1,6,15,20,15},{1,7,21,35,35},{1,8,28,56,70},{1,9,36,84,126},
 {1,10,45,120,210},{1,11,55,165,330},{1,12,66,220,495},{1,13,78,286,715},
 {1,14,91,364,1001},{1,15,105,455,1365},{1,16,120,560,1820},{1,17,136,680,2380},
 {1,18,153,816,3060},{1,19,171,969,3876}};

// m-th lexicographic 5-of-20 combination -> 20-bit membership mask
__device__ __forceinline__ unsigned unrank_mask(int m) {
  int r = m;
  unsigned mask = 0;
  int v = 0;
  #pragma unroll
  for (int t = 0; t < 5; ++t) {
    for (;; ++v) {
      int cnt = BINOM[19 - v][4 - t];   // combos with element v at slot t
      if (r < cnt) { mask |= 1u << v; ++v; break; }
      r -= cnt;
    }
  }
  return mask;
}

// One 16x16 f32 WMMA accumulation step over K=128 from an LDS tile (stride 132)
__device__ __forceinline__ v8f mm128(const float* src, const float* __restrict__ W,
                                     int c, int hf, int col) {
  v8f acc = {};
  for (int ks = 0; ks < MID / 4; ++ks) {
    const int k0 = 4 * ks;
    v2f av, bv;
    // A 16x4 f32 layout: lanes 0-15 hold K={k0,k0+1}, lanes 16-31 K={k0+2,k0+3}
    av.x = src[c * 132 + k0 + 2 * hf];
    av.y = src[c * 132 + k0 + 1 + 2 * hf];
    // B 4x16 f32 layout mirrors A: row-striped across lanes
    bv.x = W[(k0 + 2 * hf) * MID + col];
    bv.y = W[(k0 + 1 + 2 * hf) * MID + col];
    acc = __builtin_amdgcn_wmma_f32_16x16x4_f32(false, av, false, bv,
                                                (short)0, acc, false, false);
  }
  return acc;
}

// ---------------- Kernel 1: fused MLP + projection -> g = (mlp(cf)) @ Wo ----
// 80 blocks x 256 threads (8 waves). Each block: 16 rows; wave w owns N-tile w.
__global__ __launch_bounds__(256) void mlp_kernel(
    const float* __restrict__ cf,
    const float* __restrict__ W0, const float* __restrict__ bb0,
    const float* __restrict__ W1, const float* __restrict__ bb1,
    const float* __restrict__ W2, const float* __restrict__ bb2,
    const float* __restrict__ Wo, float* __restrict__ g) {
  __shared__ float a0[16 * 324];   // cf tile, stride 324 (=4 mod 64: no bank conflict)
  __shared__ float ha[16 * 132];   // h buffers, stride 132
  __shared__ float hb[16 * 132];

  const int tid  = threadIdx.x;
  const int wave = tid >> 5;
  const int lane = tid & 31;
  const int c    = lane & 15;
  const int hf   = lane >> 4;
  const int row0 = blockIdx.x * 16;
  const int col  = wave * 16 + c;

  // Stage 16x320 cf tile into LDS
  for (int i = tid; i < 16 * FEAT; i += 256) {
    const int r = i / FEAT, k = i - r * FEAT;
    a0[r * 324 + k] = cf[(row0 + r) * FEAT + k];
  }
  __syncthreads();

  // Layer 0: [16,320] @ [320,128] + b0, leaky relu -> ha
  {
    v8f acc = {};
    for (int ks = 0; ks < FEAT / 4; ++ks) {
      const int k0 = 4 * ks;
      v2f av, bv;
      av.x = a0[c * 324 + k0 + 2 * hf];
      av.y = a0[c * 324 + k0 + 1 + 2 * hf];
      bv.x = W0[(k0 + 2 * hf) * MID + col];
      bv.y = W0[(k0 + 1 + 2 * hf) * MID + col];
      acc = __builtin_amdgcn_wmma_f32_16x16x4_f32(false, av, false, bv,
                                                  (short)0, acc, false, false);
    }
    const float bias = bb0[col];
    #pragma unroll
    for (int r = 0; r < 8; ++r) {
      float x = acc[r] + bias;
      x = (x >= 0.f) ? x : 0.01f * x;
      ha[(r + 8 * hf) * 132 + col] = x;
    }
  }
  __syncthreads();

  // Layer 1: + b1, leaky relu -> hb
  {
    v8f acc = mm128(ha, W1, c, hf, col);
    const float bias = bb1[col];
    #pragma unroll
    for (int r = 0; r < 8; ++r) {
      float x = acc[r] + bias;
      x = (x >= 0.f) ? x : 0.01f * x;
      hb[(r + 8 * hf) * 132 + col] = x;
    }
  }
  __syncthreads();

  // Layer 2: + b2, no activation -> ha
  {
    v8f acc = mm128(hb, W2, c, hf, col);
    const float bias = bb2[col];
    #pragma unroll
    for (int r = 0; r < 8; ++r)
      ha[(r + 8 * hf) * 132 + col] = acc[r] + bias;
  }
  __syncthreads();

  // Layer 3: g = h2 @ Wo (no bias, no activation) -> global workspace
  {
    v8f acc = mm128(ha, Wo, c, hf, col);
    #pragma unroll
    for (int r = 0; r < 8; ++r)
      g[(row0 + r + 8 * hf) * MID + col] = acc[r];
  }
}

// ---------------- Kernel 1.5: zero per-batch softmax sums ------------------
__global__ void zero_sums(float* __restrict__ sums) { sums[threadIdx.x] = 0.f; }

// ---------------- Kernel 2: combination scoring via sparse-A WMMA ----------
// grid (64 batches, 8 groups) x 512 threads (16 waves).
// Each wave tile: D[16 combos,128] = S_tile[16,20] @ g[b][20,128] as
// 5 K-steps x 8 N-tiles of v_wmma_f32_16x16x4_f32; A built from combo bitmasks
// (value 1/5 at members), B held entirely in VGPRs.
__global__ __launch_bounds__(512) void score_kernel(
    const float* __restrict__ g, const float* __restrict__ u,
    float* __restrict__ out, float* __restrict__ sums) {
  const int b    = blockIdx.x;
  const int grp  = blockIdx.y;
  const int tid  = threadIdx.x;
  const int wave = tid >> 5;
  const int lane = tid & 31;
  const int c    = lane & 15;
  const int hf   = lane >> 4;
  __shared__ float wsum[16];

  // Preload B fragments of g[b] (20x128) and u into registers
  const float* gb = g + b * (NCAND * MID);
  v2f bf[5][8];
  float ur[8];
  #pragma unroll
  for (int n = 0; n < 8; ++n) {
    ur[n] = u[n * 16 + c];
    #pragma unroll
    for (int s = 0; s < 5; ++s) {
      bf[s][n].x = gb[(4 * s + 2 * hf) * MID + n * 16 + c];
      bf[s][n].y = gb[(4 * s + 1 + 2 * hf) * MID + n * 16 + c];
    }
  }

  float lsum = 0.f;
  const int gw = grp * 16 + wave;            // 0..127 waves per batch
  for (int t = gw; t < NTILES; t += 128) {
    const int m0 = t * 16;
    const int mm = m0 + c;                   // combo index handled by this lane
    unsigned mask = 0;
    if (mm < MCOMB) mask = unrank_mask(mm);

    v8f acc[8] = {};
    #pragma unroll
    for (int s = 0; s < 5; ++s) {
      v2f av;
      av.x = ((mask >> (4 * s + 2 * hf)) & 1u) ? 0.2f : 0.f;
      av.y = ((mask >> (4 * s + 1 + 2 * hf)) & 1u) ? 0.2f : 0.f;
      #pragma unroll
      for (int n = 0; n < 8; ++n)
        acc[n] = __builtin_amdgcn_wmma_f32_16x16x4_f32(false, av, false, bf[s][n],
                                                       (short)0, acc[n], false, false);
    }

    // relu -> dot with u -> reduce 16 columns per half-wave (butterfly)
    float part[8];
    #pragma unroll
    for (int r = 0; r < 8; ++r) {
      float p = 0.f;
      #pragma unroll
      for (int n = 0; n < 8; ++n)
        p += fmaxf(acc[n][r], 0.f) * ur[n];
      p += __shfl_xor(p, 1, 32);
      p += __shfl_xor(p, 2, 32);
      p += __shfl_xor(p, 4, 32);
      p += __shfl_xor(p, 8, 32);
      part[r] = p;
    }
    if (c == 0) {  // lanes 0 (rows 0-7) and 16 (rows 8-15)
      #pragma unroll
      for (int r = 0; r < 8; ++r) {
        const int m = m0 + r + 8 * hf;
        if (m < MCOMB) {
          const float e = expf(10.f * tanhf(part[r]));  // logits bounded: no max pass
          out[b * MCOMB + m] = e;
          lsum += e;
        }
      }
    }
  }

  // Reduce partial sums: wave butterfly, then LDS across 16 waves, atomic per batch
  #pragma unroll
  for (int off = 16; off > 0; off >>= 1) lsum += __shfl_xor(lsum, off, 32);
  if (lane == 0) wsum[wave] = lsum;
  __syncthreads();
  if (wave == 0) {
    float s = (lane < 16) ? wsum[lane] : 0.f;
    #pragma unroll
    for (int off = 8; off > 0; off >>= 1) s += __shfl_xor(s, off, 32);
    if (lane == 0) atomicAdd(&sums[b], s);
  }
}

// ---------------- Kernel 3: softmax normalization --------------------------
__global__ __launch_bounds__(256) void norm_kernel(float* __restrict__ out,
                                                   const float* __restrict__ sums) {
  const int b = blockIdx.x;
  const float inv = 1.0f / sums[b];
  for (int m = threadIdx.x; m < MCOMB; m += 256)
    out[b * MCOMB + m] *= inv;
}

extern "C" void kernel_launch(void* const* d_in, const int* in_sizes, int n_in,
                              void* d_out, int out_size, void* d_ws, size_t ws_size,
                              hipStream_t stream) {
  // setup_inputs order: batch, candidate_feature, W0, b0, W1, b1, W2, b2, Wo, u
  const float* cf = (const float*)d_in[1];
  const float* W0 = (const float*)d_in[2];
  const float* b0 = (const float*)d_in[3];
  const float* W1 = (const float*)d_in[4];
  const float* b1 = (const float*)d_in[5];
  const float* W2 = (const float*)d_in[6];
  const float* b2 = (const float*)d_in[7];
  const float* Wo = (const float*)d_in[8];
  const float* u  = (const float*)d_in[9];
  float* out = (float*)d_out;

  float* g    = (float*)d_ws;           // [1280 x 128] = 640 KB
  float* sums = g + ROWS * MID;         // [64]

  hipLaunchKernelGGL(mlp_kernel, dim3(ROWS / 16), dim3(256), 0, stream,
                     cf, W0, b0, W1, b1, W2, b2, Wo, g);
  hipLaunchKernelGGL(zero_sums, dim3(1), dim3(64), 0, stream, sums);
  hipLaunchKernelGGL(score_kernel, dim3(BATCH, 8), dim3(512), 0, stream,
                     g, u, out, sums);
  hipLaunchKernelGGL(norm_kernel, dim3(BATCH), dim3(256), 0, stream, out, sums);
}